// HiPPOMem_31164282699974
// MI455X (gfx1250) — compile-verified
//
#include <hip/hip_runtime.h>
#include <math.h>

// Problem constants (from reference)
#define BATCH   32
#define D_IN    128
#define TLEN    4096
#define HIDDEN  256
#define FUTURE  128
#define NORD    16
#define CHUNK   256   // steps per recurrence launch

typedef float v2f __attribute__((ext_vector_type(2)));
typedef float v8f __attribute__((ext_vector_type(8)));

// ---------------------------------------------------------------------------
// Kernel 0: HiPPO-LegS bilinear transition matrices.
// P = I + dt/2*A, Q = I - dt/2*A are lower triangular (A_legs is lower tri),
// so A_t = P^{-1}Q and B_t = dt*P^{-1}Bv via forward substitution. One thread
// per timestep. Also zero-initializes recurrent state (ws is poisoned 0xAA).
// ---------------------------------------------------------------------------
__global__ __launch_bounds__(256)
void hippo_transitions(float* __restrict__ At, float* __restrict__ Bt,
                       float* __restrict__ hState, float* __restrict__ cState) {
    const int t0 = blockIdx.x * 256 + threadIdx.x;   // 0..4095
    if (t0 < TLEN) {
        const float dt  = 1.0f / (float)(t0 + 1);
        const float hdt = 0.5f * dt;
        float* Arow = At + (size_t)t0 * (NORD * NORD);
        for (int j = 0; j < NORD; ++j) {
            const float rj = sqrtf(2.0f * j + 1.0f);
            for (int i = 0; i < j; ++i) Arow[i * NORD + j] = 0.0f;
            const float yj = (1.0f - hdt * (j + 1)) / (1.0f + hdt * (j + 1));
            Arow[j * NORD + j] = yj;
            float run = rj * yj;                       // sum_{k=j..i-1} r_k y_k
            for (int i = j + 1; i < NORD; ++i) {
                const float ri = sqrtf(2.0f * i + 1.0f);
                const float yi = -hdt * ri * (rj + run) / (1.0f + hdt * (i + 1));
                Arow[i * NORD + j] = yi;
                run += ri * yi;
            }
        }
        float run = 0.0f;                              // sum r_k b_k
        for (int i = 0; i < NORD; ++i) {
            const float ri = sqrtf(2.0f * i + 1.0f);
            const float bi = ri * (dt - hdt * run) / (1.0f + hdt * (i + 1));
            Bt[t0 * NORD + i] = bi;
            run += ri * bi;
        }
    }
    const int g = blockIdx.x * 256 + threadIdx.x;      // 4096 threads total
    for (int i = g; i < BATCH * HIDDEN; i += 4096) hState[i] = 0.0f;
    for (int i = g; i < BATCH * NORD;   i += 4096) cState[i] = 0.0f;
}

// ---------------------------------------------------------------------------
// Kernel 1: state-independent x-projections for one chunk of timesteps.
// One workgroup per timestep t: computes x_t @ [Wxh;Wxg]^T (32x512, K=128)
// with V_WMMA_F32_16X16X4_F32, plus the scalar HiPPO feature x-part.
// The x_t column gather (stride T floats) is staged memory->LDS with
// GLOBAL_LOAD_ASYNC_TO_LDS_B32 (ASYNCcnt path), no VGPR round trip.
// ---------------------------------------------------------------------------
__global__ __launch_bounds__(256)
void hippo_xproj(const float* __restrict__ ts, const float* __restrict__ Wxh,
                 const float* __restrict__ Wxg, const float* __restrict__ Wu,
                 float* __restrict__ PH, float* __restrict__ PG,
                 float* __restrict__ PF, int t0) {
    __shared__ float xt[BATCH * 132];
    const int tid = threadIdx.x;
    const int t   = t0 + blockIdx.x;
    // async gather of x_t (ts layout (B, D_IN, T): element stride T in (b,d))
    for (int i = tid; i < BATCH * D_IN; i += 256) {
        const int bb = i >> 7, d = i & 127;
        const unsigned ldsoff = (unsigned)(size_t)&xt[bb * 132 + d];
        const unsigned goff   = (unsigned)((((size_t)bb * D_IN + d) * TLEN + t)
                                           * sizeof(float));
        asm volatile("global_load_async_to_lds_b32 %0, %1, %2"
                     :: "v"(ldsoff), "v"(goff), "s"(ts)
                     : "memory");
    }
    asm volatile("s_wait_asynccnt 0" ::: "memory");
    __syncthreads();

    const int lane  = tid & 31;
    const int wave  = tid >> 5;
    const int lm    = lane & 15;
    const int kbase = (lane < 16) ? 0 : 2;   // A/B f32 K striping
    const int s     = t - t0;

    for (int w8 = 0; w8 < 8; ++w8) {
        const int tt = wave + 8 * w8;        // 64 output tiles (2 x 32)
        const int mt = tt & 1;
        const int nc = tt >> 1;
        const int m  = 16 * mt + lm;
        const int o  = 16 * nc + lm;         // output channel 0..511
        const float* Wrow = (o < HIDDEN) ? (Wxh + (size_t)o * D_IN)
                                         : (Wxg + (size_t)(o - HIDDEN) * D_IN);
        const float* arow = &xt[m * 132];
        v8f acc = {};
        for (int k0 = 0; k0 < D_IN; k0 += 4) {
            v2f a, bv;
            a.x  = arow[k0 + kbase];  a.y  = arow[k0 + kbase + 1];
            bv.x = Wrow[k0 + kbase];  bv.y = Wrow[k0 + kbase + 1];
            acc = __builtin_amdgcn_wmma_f32_16x16x4_f32(
                false, a, false, bv, (short)0, acc, false, false);
        }
        float* dst = (o < HIDDEN) ? (PH + (size_t)s * BATCH * HIDDEN)
                                  : (PG + (size_t)s * BATCH * HIDDEN);
        const int oc = (o < HIDDEN) ? o : o - HIDDEN;
        for (int v = 0; v < 8; ++v) {
            const int brow = 16 * mt + v + ((lane < 16) ? 0 : 8);
            dst[(size_t)brow * HIDDEN + oc] = acc[v];
        }
    }
    __syncthreads();
    if (tid < BATCH) {                        // f x-part: x_t . Wu[0, :128]
        float acc = 0.0f;
        for (int d = 0; d < D_IN; ++d) acc += xt[tid * 132 + d] * Wu[d];
        PF[s * BATCH + tid] = acc;
    }
}

// ---------------------------------------------------------------------------
// Kernel 2: sequential recurrence for one chunk, single workgroup (8 waves).
// State (h, c) lives in LDS across the chunk. Per step:
//   (a) all 256 threads: A_t/B_t stage (coalesced) + 8-way-split h.Wu_h
//       partials (no idle waves on the critical path),
//   (b) HiPPO update c_t = A_t c + f B_t (2 entries/thread, folds partials),
//   (c) fused WMMA GEMM [h|c](32x272) x (272x512) -> sH|sG pre-activations,
//   (d) elementwise gate/tanh combine overwriting h.
// LDS strides 273 / 516 chosen conflict-free (gcd(17,64)=1).
// ---------------------------------------------------------------------------
__global__ __launch_bounds__(256)
void hippo_recur(const float* __restrict__ At, const float* __restrict__ Bt,
                 const float* __restrict__ PH, const float* __restrict__ PG,
                 const float* __restrict__ PF, const float* __restrict__ Wu,
                 const float* __restrict__ Whh, const float* __restrict__ Wmh,
                 const float* __restrict__ Whg, const float* __restrict__ Wmg,
                 const float* __restrict__ bhp, const float* __restrict__ bgp,
                 const float* __restrict__ bup,
                 float* __restrict__ hState, float* __restrict__ cState,
                 int t0, int nsteps) {
    __shared__ float hc[BATCH * 273];    // cols 0..255 = h, 256..271 = c
    __shared__ float spre[BATCH * 516];  // cols 0..255 = sH, 256..511 = sG
    __shared__ float wuh[HIDDEN], bhs[HIDDEN], bgs[HIDDEN];
    __shared__ float fpart[BATCH * 8];   // 8-way split of h . Wu_h per batch
    __shared__ float Ats[NORD * NORD], Bts[NORD];
    __shared__ float bus;

    const int tid = threadIdx.x;
    for (int i = tid; i < BATCH * HIDDEN; i += 256)
        hc[(i >> 8) * 273 + (i & 255)] = hState[i];
    for (int i = tid; i < BATCH * NORD; i += 256)
        hc[(i >> 4) * 273 + 256 + (i & 15)] = cState[i];
    wuh[tid] = Wu[D_IN + tid];
    bhs[tid] = bhp[tid];
    bgs[tid] = bgp[tid];
    if (tid == 0) bus = bup[0];
    __syncthreads();

    const int lane  = tid & 31;
    const int wave  = tid >> 5;
    const int lm    = lane & 15;
    const int kbase = (lane < 16) ? 0 : 2;
    const int cb    = tid >> 3;              // batch row for f/c work
    const int seg   = tid & 7;               // 32-wide segment of the f dot
    const int ci    = seg * 2;               // two c entries per thread

    for (int s = 0; s < nsteps; ++s) {
        const int tg = t0 + s;
        // ---- (a) stage A_t/B_t (coalesced) + f-dot partials ----
        Ats[tid & 255] = At[(size_t)tg * (NORD * NORD) + tid];
        if (tid < NORD) Bts[tid] = Bt[tg * NORD + tid];
        {
            const float* hr = &hc[cb * 273 + seg * 32];
            const float* wr = &wuh[seg * 32];
            float p = 0.0f;
            for (int o = 0; o < 32; ++o) p += hr[o] * wr[o];
            fpart[tid] = p;
        }
        __syncthreads();
        // ---- (b) c_t = A_t c_{t-1} + f * B_t ----
        float cn0, cn1;
        {
            float fb = PF[s * BATCH + cb] + bus;
            for (int k = 0; k < 8; ++k) fb += fpart[cb * 8 + k];
            const float* Ar0 = &Ats[ci * NORD];
            const float* Ar1 = Ar0 + NORD;
            const float* cr  = &hc[cb * 273 + 256];
            float a0 = 0.0f, a1 = 0.0f;
            for (int j = 0; j < NORD; ++j) {
                const float cj = cr[j];
                a0 += Ar0[j] * cj;
                a1 += Ar1[j] * cj;
            }
            cn0 = a0 + fb * Bts[ci];
            cn1 = a1 + fb * Bts[ci + 1];
        }
        __syncthreads();
        hc[cb * 273 + 256 + ci]     = cn0;
        hc[cb * 273 + 256 + ci + 1] = cn1;
        __syncthreads();
        // ---- (c) fused recurrent GEMM: [h|c] x [Whh|Wmh ; Whg|Wmg]^T ----
        for (int w8 = 0; w8 < 8; ++w8) {
            const int tt = wave + 8 * w8;    // 64 tiles (2 x 32)
            const int mt = tt & 1;
            const int nc = tt >> 1;
            const int m  = 16 * mt + lm;
            const int n  = 16 * nc + lm;     // 0..511
            const int gsel = (n >= HIDDEN);
            const int nn   = gsel ? (n - HIDDEN) : n;
            const float* Wh = gsel ? (Whg + (size_t)nn * HIDDEN)
                                   : (Whh + (size_t)nn * HIDDEN);
            const float* Wm = gsel ? (Wmg + (size_t)nn * NORD)
                                   : (Wmh + (size_t)nn * NORD);
            const float* arow = &hc[m * 273];
            v8f acc = {};
            for (int k0 = 0; k0 < HIDDEN + NORD; k0 += 4) {
                const int k = k0 + kbase;
                v2f a, bv;
                a.x  = arow[k];
                a.y  = arow[k + 1];
                bv.x = (k     < HIDDEN) ? Wh[k]     : Wm[k - HIDDEN];
                bv.y = (k + 1 < HIDDEN) ? Wh[k + 1] : Wm[k + 1 - HIDDEN];
                acc = __builtin_amdgcn_wmma_f32_16x16x4_f32(
                    false, a, false, bv, (short)0, acc, false, false);
            }
            for (int v = 0; v < 8; ++v) {
                const int brow = 16 * mt + v + ((lane < 16) ? 0 : 8);
                spre[brow * 516 + n] = acc[v];
            }
        }
        __syncthreads();
        // ---- (d) gate / candidate combine, overwrite h in LDS ----
        for (int e = tid; e < BATCH * HIDDEN; e += 256) {
            const int eb = e >> 8, o = e & 255;
            const size_t p = ((size_t)s * BATCH + eb) * HIDDEN + o;
            const float sh = spre[eb * 516 + o]       + PH[p] + bhs[o];
            const float sg = spre[eb * 516 + 256 + o] + PG[p] + bgs[o];
            const float gt = 1.0f / (1.0f + expf(-sg));
            const float ht = tanhf(sh);
            const float ho = hc[eb * 273 + o];
            hc[eb * 273 + o] = (1.0f - gt) * ho + gt * ht;
        }
        __syncthreads();
    }
    for (int i = tid; i < BATCH * HIDDEN; i += 256)
        hState[i] = hc[(i >> 8) * 273 + (i & 255)];
    for (int i = tid; i < BATCH * NORD; i += 256)
        cState[i] = hc[(i >> 4) * 273 + 256 + (i & 15)];
}

// ---------------------------------------------------------------------------
// Kernel 3: readout  out = h @ Wout^T + bout   (tiny: 32x128, K=256)
// ---------------------------------------------------------------------------
__global__ __launch_bounds__(256)
void hippo_readout(const float* __restrict__ hState,
                   const float* __restrict__ Wout,
                   const float* __restrict__ bout, float* __restrict__ out) {
    const int e  = blockIdx.x * 256 + threadIdx.x;  // 4096 = 32*128
    const int b  = e >> 7;
    const int fo = e & 127;
    float acc = bout[fo];
    const float* wr = Wout + (size_t)fo * HIDDEN;
    const float* hr = hState + (size_t)b * HIDDEN;
    for (int o = 0; o < HIDDEN; ++o) acc += hr[o] * wr[o];
    out[e] = acc;
}

// ---------------------------------------------------------------------------
extern "C" void kernel_launch(void* const* d_in, const int* in_sizes, int n_in,
                              void* d_out, int out_size, void* d_ws, size_t ws_size,
                              hipStream_t stream) {
    const float* ts   = (const float*)d_in[0];
    const float* Wu   = (const float*)d_in[1];
    const float* bu   = (const float*)d_in[2];
    const float* Wxh  = (const float*)d_in[3];
    const float* Whh  = (const float*)d_in[4];
    const float* Wmh  = (const float*)d_in[5];
    const float* bh   = (const float*)d_in[6];
    const float* Wxg  = (const float*)d_in[7];
    const float* Whg  = (const float*)d_in[8];
    const float* Wmg  = (const float*)d_in[9];
    const float* bg   = (const float*)d_in[10];
    const float* Wout = (const float*)d_in[11];
    const float* bout = (const float*)d_in[12];

    float* ws = (float*)d_ws;                       // ~21.3 MB total
    float* At = ws;                                 // 4096*256
    float* Bt = At + (size_t)TLEN * NORD * NORD;    // 4096*16
    float* hS = Bt + (size_t)TLEN * NORD;           // 32*256
    float* cS = hS + BATCH * HIDDEN;                // 32*16
    float* PH = cS + BATCH * NORD;                  // CHUNK*32*256
    float* PG = PH + (size_t)CHUNK * BATCH * HIDDEN;
    float* PF = PG + (size_t)CHUNK * BATCH * HIDDEN;

    hippo_transitions<<<16, 256, 0, stream>>>(At, Bt, hS, cS);
    for (int t0 = 0; t0 < TLEN; t0 += CHUNK) {
        hippo_xproj<<<CHUNK, 256, 0, stream>>>(ts, Wxh, Wxg, Wu, PH, PG, PF, t0);
        hippo_recur<<<1, 256, 0, stream>>>(At, Bt, PH, PG, PF, Wu,
                                           Whh, Wmh, Whg, Wmg, bh, bg, bu,
                                           hS, cS, t0, CHUNK);
    }
    hippo_readout<<<16, 256, 0, stream>>>(hS, Wout, bout, (float*)d_out);
}